// Mirror3d_FastRCNNOutputLayers_23639499997197
// MI455X (gfx1250) — compile-verified
//
#include <hip/hip_runtime.h>
#include <stdint.h>

#define R_N 4096
#define NA 8
#define TOPK_N 100
#define IMG_W_F 1333.0f
#define IMG_H_F 800.0f
#define SCORE_TH 0.05f
#define NMS_TH 0.5f
#define SUP_WORDS 128   // 4096 / 32

typedef __attribute__((ext_vector_type(2))) float v2f;
typedef __attribute__((ext_vector_type(8))) float v8f;

// ---------------------------------------------------------------------------
// K1: per-row prep: clip boxes, cls score, anchor argmax, residual gather
// ---------------------------------------------------------------------------
__global__ __launch_bounds__(256) void prep_kernel(
    const float* __restrict__ boxes, const float* __restrict__ scores,
    const float* __restrict__ ascores, const float* __restrict__ resid,
    float* __restrict__ boxc, float* __restrict__ cls,
    int* __restrict__ aidx, float* __restrict__ asc, float* __restrict__ res3) {
  int r = blockIdx.x * blockDim.x + threadIdx.x;
  if (r >= R_N) return;
  float x1 = fminf(fmaxf(boxes[r * 4 + 0], 0.f), IMG_W_F);
  float y1 = fminf(fmaxf(boxes[r * 4 + 1], 0.f), IMG_H_F);
  float x2 = fminf(fmaxf(boxes[r * 4 + 2], 0.f), IMG_W_F);
  float y2 = fminf(fmaxf(boxes[r * 4 + 3], 0.f), IMG_H_F);
  boxc[r * 4 + 0] = x1; boxc[r * 4 + 1] = y1;
  boxc[r * 4 + 2] = x2; boxc[r * 4 + 3] = y2;
  cls[r] = scores[r * 2 + 0];
  // argmax over 8 anchor scores, first max wins (JAX semantics)
  int best = 0;
  float bv = ascores[r * NA + 0];
#pragma unroll
  for (int a = 1; a < NA; ++a) {
    float v = ascores[r * NA + a];
    if (v > bv) { bv = v; best = a; }
  }
  aidx[r] = best;
  asc[r] = bv;
  res3[r * 3 + 0] = resid[r * NA * 3 + best * 3 + 0];
  res3[r * 3 + 1] = resid[r * NA * 3 + best * 3 + 1];
  res3[r * 3 + 2] = resid[r * NA * 3 + best * 3 + 2];
}

// ---------------------------------------------------------------------------
// K2: single-block bitonic sort of 4096 scores (descending) carrying index
// ---------------------------------------------------------------------------
__global__ __launch_bounds__(1024) void sort_kernel(
    const float* __restrict__ cls, float* __restrict__ s_score,
    int* __restrict__ s_order) {
  __shared__ float kk[R_N];
  __shared__ int vv[R_N];
  const int tid = threadIdx.x;
  for (int i = tid; i < R_N; i += 1024) { kk[i] = cls[i]; vv[i] = i; }
  __syncthreads();
  for (int len = 2; len <= R_N; len <<= 1) {
    for (int j = len >> 1; j > 0; j >>= 1) {
      for (int i = tid; i < R_N; i += 1024) {
        int ixj = i ^ j;
        if (ixj > i) {
          bool up = ((i & len) == 0);
          float a = kk[i], b = kk[ixj];
          bool sw = up ? (a < b) : (a > b);  // descending overall
          if (sw) {
            kk[i] = b; kk[ixj] = a;
            int t = vv[i]; vv[i] = vv[ixj]; vv[ixj] = t;
          }
        }
      }
      __syncthreads();
    }
  }
  for (int i = tid; i < R_N; i += 1024) { s_score[i] = kk[i]; s_order[i] = vv[i]; }
}

// ---------------------------------------------------------------------------
// K2b: gather boxes into sorted order + areas
// ---------------------------------------------------------------------------
__global__ __launch_bounds__(256) void gather_kernel(
    const float* __restrict__ boxc, const int* __restrict__ s_order,
    float* __restrict__ sb, float* __restrict__ sarea) {
  int i = blockIdx.x * blockDim.x + threadIdx.x;
  if (i >= R_N) return;
  int o = s_order[i];
  float x1 = boxc[o * 4 + 0], y1 = boxc[o * 4 + 1];
  float x2 = boxc[o * 4 + 2], y2 = boxc[o * 4 + 3];
  sb[i * 4 + 0] = x1; sb[i * 4 + 1] = y1;
  sb[i * 4 + 2] = x2; sb[i * 4 + 3] = y2;
  sarea[i] = (x2 - x1) * (y2 - y1);
}

// ---------------------------------------------------------------------------
// K3: suppression bitmask matrix.  Union denominator base (area_i + area_j)
// computed as a rank-2 outer product on the WMMA pipe (V_WMMA_F32_16X16X4_F32);
// intersection on VALU.  Block = 8 wave32s; block tile = 16 rows x 256 cols;
// each wave produces one 32-bit mask word per row (two 16x16 WMMA tiles).
// ---------------------------------------------------------------------------
__global__ __launch_bounds__(256) void iou_mask_kernel(
    const float* __restrict__ sb, const float* __restrict__ sarea,
    uint32_t* __restrict__ sup) {
  __shared__ float rbox[16][4];
  __shared__ float rarea[16];
  __shared__ float cbox[256][4];
  __shared__ float carea[256];
  __shared__ unsigned char bits[8][16][32];

  const int tid = threadIdx.x;
  const int rb = blockIdx.x;        // 16-row block
  const int cb = blockIdx.y;        // 256-col superblock
  const int row0 = rb * 16;
  const int col0 = cb * 256;

  if (tid < 16) {
    int gr = row0 + tid;
    rbox[tid][0] = sb[gr * 4 + 0];
    rbox[tid][1] = sb[gr * 4 + 1];
    rbox[tid][2] = sb[gr * 4 + 2];
    rbox[tid][3] = sb[gr * 4 + 3];
    rarea[tid] = sarea[gr];
  }
  {
    int gc = col0 + tid;
    cbox[tid][0] = sb[gc * 4 + 0];
    cbox[tid][1] = sb[gc * 4 + 1];
    cbox[tid][2] = sb[gc * 4 + 2];
    cbox[tid][3] = sb[gc * 4 + 3];
    carea[tid] = sarea[gc];
  }
  __syncthreads();

  const int wave = tid >> 5;
  const int lane = tid & 31;
  const int nloc = lane & 15;
  const int mbase = (lane < 16) ? 0 : 8;

  // A fragment (16x4 f32): A[m][0] = area_row[m], A[m][1] = 1, K=2..3 zero.
  v2f afr;
  if (lane < 16) { afr.x = rarea[lane]; afr.y = 1.0f; }
  else           { afr.x = 0.0f;        afr.y = 0.0f; }

#pragma unroll
  for (int t = 0; t < 2; ++t) {
    const int cl = wave * 32 + t * 16 + nloc;   // local col within 256
    // B fragment (4x16 f32): B[0][n] = 1, B[1][n] = area_col[n], K=2..3 zero.
    v2f bfr;
    if (lane < 16) { bfr.x = 1.0f; bfr.y = carea[cl]; }
    else           { bfr.x = 0.0f; bfr.y = 0.0f; }
    v8f acc = {};
    acc = __builtin_amdgcn_wmma_f32_16x16x4_f32(
        /*neg_a=*/false, afr, /*neg_b=*/false, bfr,
        /*c_mod=*/(short)0, acc, /*reuse_a=*/false, /*reuse_b=*/false);

    const float cx1 = cbox[cl][0], cy1 = cbox[cl][1];
    const float cx2 = cbox[cl][2], cy2 = cbox[cl][3];
    const int gcol = col0 + cl;
#pragma unroll
    for (int k = 0; k < 8; ++k) {
      const int m = mbase + k;                  // D layout: lanes<16 -> M=k, else M=k+8
      float ix1 = fmaxf(rbox[m][0], cx1);
      float iy1 = fmaxf(rbox[m][1], cy1);
      float ix2 = fminf(rbox[m][2], cx2);
      float iy2 = fminf(rbox[m][3], cy2);
      float iw = fmaxf(ix2 - ix1, 0.0f);
      float ih = fmaxf(iy2 - iy1, 0.0f);
      float inter = iw * ih;
      float iou = inter / (acc[k] - inter + 1e-12f);   // acc[k] = area_m + area_n
      int grow = row0 + m;
      bits[wave][m][t * 16 + nloc] =
          (iou > NMS_TH && gcol > grow) ? (unsigned char)1 : (unsigned char)0;
    }
  }
  __syncthreads();

  if (tid < 128) {
    int w = tid >> 4, m = tid & 15;
    uint32_t word = 0;
#pragma unroll
    for (int n = 0; n < 32; ++n)
      word |= ((uint32_t)bits[w][m][n]) << n;
    sup[(size_t)(row0 + m) * SUP_WORDS + (size_t)(cb * 8 + w)] = word;
  }
}

// ---------------------------------------------------------------------------
// K4: sequential greedy scan over sorted rows (bitmask propagation) + output
// ---------------------------------------------------------------------------
__global__ __launch_bounds__(128) void nms_out_kernel(
    const float* __restrict__ s_score, const int* __restrict__ s_order,
    const float* __restrict__ sb, const uint32_t* __restrict__ sup,
    const int* __restrict__ aidx, const float* __restrict__ asc,
    const float* __restrict__ res3, float* __restrict__ out) {
  __shared__ uint32_t removed[SUP_WORDS];
  __shared__ int keptlist[TOPK_N];
  __shared__ int count;
  __shared__ int flag;
  const int tid = threadIdx.x;
  removed[tid] = 0u;
  if (tid == 0) { count = 0; }
  __syncthreads();

  for (int i = 0; i < R_N; ++i) {
    if (tid == 0) {
      flag = 0;
      bool alive = (s_score[i] > SCORE_TH) &&
                   !((removed[i >> 5] >> (i & 31)) & 1u);
      if (alive) { keptlist[count++] = i; flag = 1; }
    }
    __syncthreads();
    if (count >= TOPK_N) break;          // uniform: shared, read post-barrier
    if (flag) removed[tid] |= sup[(size_t)i * SUP_WORDS + tid];
    __syncthreads();
  }
  __syncthreads();

  // Output layout (floats): out9[100*9] | pred_classes[100] |
  //                         pred_anchor_classes[100] | kept_row_inds[100]
  for (int t = tid; t < TOPK_N; t += 128) {
    float* row = out + t * 9;
    if (t < count) {
      int i = keptlist[t];
      int o = s_order[i];
      row[0] = sb[i * 4 + 0]; row[1] = sb[i * 4 + 1];
      row[2] = sb[i * 4 + 2]; row[3] = sb[i * 4 + 3];
      row[4] = s_score[i];
      row[5] = asc[o];
      row[6] = res3[o * 3 + 0];
      row[7] = res3[o * 3 + 1];
      row[8] = res3[o * 3 + 2];
      out[900 + t]  = 0.0f;
      out[1000 + t] = (float)aidx[o];
      out[1100 + t] = (float)o;
    } else {
#pragma unroll
      for (int q = 0; q < 9; ++q) row[q] = 0.0f;
      out[900 + t]  = -1.0f;
      out[1000 + t] = -1.0f;
      out[1100 + t] = -1.0f;
    }
  }
}

// ---------------------------------------------------------------------------
extern "C" void kernel_launch(void* const* d_in, const int* in_sizes, int n_in,
                              void* d_out, int out_size, void* d_ws, size_t ws_size,
                              hipStream_t stream) {
  (void)in_sizes; (void)n_in; (void)out_size; (void)ws_size;
  const float* boxes   = (const float*)d_in[0];   // (4096,4)
  const float* scores  = (const float*)d_in[1];   // (4096,2)
  const float* ascores = (const float*)d_in[2];   // (4096,8)
  const float* resid   = (const float*)d_in[3];   // (4096,24)
  float* out = (float*)d_out;                     // 1200 floats

  char* base = (char*)d_ws;
  size_t off = 0;
  auto alloc = [&](size_t bytes) {
    char* p = base + off;
    off = (off + bytes + 255) & ~(size_t)255;
    return p;
  };
  float*    boxc    = (float*)alloc(R_N * 4 * sizeof(float));
  float*    cls     = (float*)alloc(R_N * sizeof(float));
  int*      aidx    = (int*)alloc(R_N * sizeof(int));
  float*    asc     = (float*)alloc(R_N * sizeof(float));
  float*    res3    = (float*)alloc(R_N * 3 * sizeof(float));
  float*    s_score = (float*)alloc(R_N * sizeof(float));
  int*      s_order = (int*)alloc(R_N * sizeof(int));
  float*    sb      = (float*)alloc(R_N * 4 * sizeof(float));
  float*    sarea   = (float*)alloc(R_N * sizeof(float));
  uint32_t* sup     = (uint32_t*)alloc((size_t)R_N * SUP_WORDS * sizeof(uint32_t));

  prep_kernel<<<R_N / 256, 256, 0, stream>>>(boxes, scores, ascores, resid,
                                             boxc, cls, aidx, asc, res3);
  sort_kernel<<<1, 1024, 0, stream>>>(cls, s_score, s_order);
  gather_kernel<<<R_N / 256, 256, 0, stream>>>(boxc, s_order, sb, sarea);
  iou_mask_kernel<<<dim3(R_N / 16, R_N / 256), 256, 0, stream>>>(sb, sarea, sup);
  nms_out_kernel<<<1, 128, 0, stream>>>(s_score, s_order, sb, sup,
                                        aidx, asc, res3, out);
}